// InversePINO_16801912062492
// MI455X (gfx1250) — compile-verified
//
#include <hip/hip_runtime.h>
#include <hip/hip_bf16.h>
#include <math.h>

// ---------------------------------------------------------------------------
// FNO-3D forward for MI455X (gfx1250, wave32).
// Strategy: truncated separable DFTs as GEMMs; f32 WMMA (16x16x4) on the
// three 128MB-traffic GEMMs; VALU+LDS twiddles for the tiny middle stages.
// ---------------------------------------------------------------------------

typedef __attribute__((ext_vector_type(2))) float v2f;
typedef __attribute__((ext_vector_type(8))) float v8f;

#define PI_F 3.14159265358979323846f
#define CH   32
#define DHW  262144L            // 64^3
#define RROWS 524288L           // B*C*64*64

__device__ __forceinline__ v8f wmma_f32_16x16x4(v2f a, v2f b, v8f c) {
#if defined(__has_builtin) && __has_builtin(__builtin_amdgcn_wmma_f32_16x16x4_f32)
  return __builtin_amdgcn_wmma_f32_16x16x4_f32(false, a, false, b, (short)0, c, false, false);
#else
  // fallback marker (should not trigger on gfx1250): keeps compile alive.
  c[0] += a.x * b.x + a.y * b.y;
  return c;
#endif
}

__device__ __forceinline__ float gelu_tanh(float x) {
  float x3 = x * x * x;
  float u = 0.7978845608028654f * (x + 0.044715f * x3);
  return 0.5f * x * (1.0f + tanhf(u));
}

// ---------------------------------------------------------------------------
// Lift: x[b,c,p] = v*lw[0,c] + t*lw[1,c] + lb[c].  1,048,576 threads.
// ---------------------------------------------------------------------------
__global__ __launch_bounds__(256) void lift_kernel(
    const float* __restrict__ vel, const float* __restrict__ ptt,
    const float* __restrict__ lw, const float* __restrict__ lb,
    float* __restrict__ X) {
  long p = (long)blockIdx.x * 256 + threadIdx.x;     // over B*DHW
  int b = (int)(p >> 18);
  long sp = p & (DHW - 1);
  float a = vel[p], c = ptt[p];
  long base = (long)b * CH * DHW + sp;
  for (int co = 0; co < CH; ++co)
    X[base + (long)co * DHW] = a * lw[co] + c * lw[CH + co] + lb[co];
}

// ---------------------------------------------------------------------------
// Forward DFT along W (WMMA GEMM): [RROWS,64] x [64,16] -> F1[RROWS,16]
// cols 0..7 = Re(kw), 8..15 = Im(kw) of exp(-i 2pi kw w/64).
// Block = 8 waves; each wave owns one 16x16 output tile, K=64 (16 wmma steps).
// ---------------------------------------------------------------------------
__global__ __launch_bounds__(256) void dftw_fwd(
    const float* __restrict__ X, float* __restrict__ F1) {
  __shared__ float Bm[64 * 16];                      // Bm[k*16+n]
  int t = threadIdx.x;
  for (int i = t; i < 1024; i += 256) {
    int k = i >> 4, n = i & 15;
    float ang = 2.f * PI_F * (float)((n & 7) * k) / 64.f;
    Bm[i] = (n < 8) ? __cosf(ang) : -__sinf(ang);
  }
  __syncthreads();
  int lane = t & 31, wv = t >> 5;
  long tile = (long)blockIdx.x * 8 + wv;             // 32768 tiles
  long row0 = tile * 16;
  const float* Arow = X + row0 * 64;
  int m = lane & 15;
  int koff = (lane >> 4) << 1;                       // 0 or 2
  int mhi  = (lane >> 4) << 3;                       // 0 or 8
  v8f acc = {0.f,0.f,0.f,0.f,0.f,0.f,0.f,0.f};
  for (int kk = 0; kk < 64; kk += 4) {
    int ka = kk + koff;
    v2f a; a.x = Arow[m * 64 + ka];      a.y = Arow[m * 64 + ka + 1];
    v2f b; b.x = Bm[ka * 16 + m];        b.y = Bm[(ka + 1) * 16 + m];
    acc = wmma_f32_16x16x4(a, b, acc);
  }
  float* Orow = F1 + row0 * 16;
  for (int r = 0; r < 8; ++r) Orow[(r + mhi) * 16 + m] = acc[r];
}

// ---------------------------------------------------------------------------
// Forward DFT along H: F1[(b,c,d),64,16] -> F2[(b,c,d),16,16] (complex e^-i).
// 1,048,576 threads, LDS twiddle table.
// ---------------------------------------------------------------------------
__global__ __launch_bounds__(256) void dfth_fwd(
    const float* __restrict__ F1, float* __restrict__ F2) {
  __shared__ float cs[1024], sn[1024];               // [kh(16)][h(64)]
  int t = threadIdx.x;
  for (int i = t; i < 1024; i += 256) {
    int kh = i >> 6, h = i & 63;
    int f = (kh < 8) ? kh : (48 + kh);               // 56..63
    float ang = 2.f * PI_F * (float)(f * h) / 64.f;
    cs[i] = __cosf(ang); sn[i] = __sinf(ang);
  }
  __syncthreads();
  long gid = (long)blockIdx.x * 256 + t;             // 2^20 total
  int kw = (int)(gid & 7);
  int kh = (int)((gid >> 3) & 15);
  long bcd = gid >> 7;
  const float* base = F1 + bcd * 1024;
  __builtin_prefetch(base, 0, 0);
  float orr = 0.f, oii = 0.f;
  for (int h = 0; h < 64; ++h) {
    float r = base[h * 16 + kw], s = base[h * 16 + 8 + kw];
    float c = cs[kh * 64 + h], si = sn[kh * 64 + h];
    orr += r * c + s * si;                           // e^{-i}
    oii += s * c - r * si;
  }
  float* ob = F2 + bcd * 256;
  ob[kh * 16 + kw] = orr;
  ob[kh * 16 + 8 + kw] = oii;
}

// ---------------------------------------------------------------------------
// Forward DFT along D: F2[(b,c),64,16,16] -> SP[(b,c),16,16,16]. 262,144 thr.
// ---------------------------------------------------------------------------
__global__ __launch_bounds__(256) void dftd_fwd(
    const float* __restrict__ F2, float* __restrict__ SP) {
  __shared__ float cs[1024], sn[1024];               // [kd(16)][d(64)]
  int t = threadIdx.x;
  for (int i = t; i < 1024; i += 256) {
    int kd = i >> 6, d = i & 63;
    int f = (kd < 8) ? kd : (48 + kd);
    float ang = 2.f * PI_F * (float)(f * d) / 64.f;
    cs[i] = __cosf(ang); sn[i] = __sinf(ang);
  }
  __syncthreads();
  long gid = (long)blockIdx.x * 256 + t;             // 2^18
  int kw = (int)(gid & 7);
  int kh = (int)((gid >> 3) & 15);
  int kd = (int)((gid >> 7) & 15);
  long bc = gid >> 11;
  const float* base = F2 + bc * 16384 + kh * 16 + kw;
  float orr = 0.f, oii = 0.f;
  for (int d = 0; d < 64; ++d) {
    float r = base[d * 256], s = base[d * 256 + 8];
    float c = cs[kd * 64 + d], si = sn[kd * 64 + d];
    orr += r * c + s * si;
    oii += s * c - r * si;
  }
  float* ob = SP + bc * 4096 + (kd * 16 + kh) * 16 + kw;
  ob[0] = orr; ob[8] = oii;
}

// ---------------------------------------------------------------------------
// Spectral core: per mode point, complex [4x32]x[32x32]. 2048 blocks x 128.
// ---------------------------------------------------------------------------
__global__ __launch_bounds__(128) void spectral_core(
    const float* __restrict__ SP, const float* __restrict__ wr,
    const float* __restrict__ wi, float* __restrict__ SPo) {
  int point = blockIdx.x;                            // kd*128 + kh*8 + kw
  int kw = point & 7, kh = (point >> 3) & 15, kd = (point >> 7) & 15;
  int corner = ((kd >= 8) ? 2 : 0) | ((kh >= 8) ? 1 : 0);
  int xd = kd & 7, xh = kh & 7;
  int t = threadIdx.x;
  int co = t & 31, b = t >> 5;
  long wbase = (long)corner * 524288 + (long)co * 512 + xd * 64 + xh * 8 + kw;
  const float* spb = SP + (long)b * CH * 4096 + (kd * 16 + kh) * 16 + kw;
  float orr = 0.f, oii = 0.f;
  for (int ci = 0; ci < CH; ++ci) {
    float zr = spb[(long)ci * 4096], zi = spb[(long)ci * 4096 + 8];
    float wrv = wr[wbase + (long)ci * 16384];
    float wiv = wi[wbase + (long)ci * 16384];
    orr += zr * wrv - zi * wiv;
    oii += zr * wiv + zi * wrv;
  }
  float* ob = SPo + ((long)b * CH + co) * 4096 + (kd * 16 + kh) * 16 + kw;
  ob[0] = orr; ob[8] = oii;
}

// ---------------------------------------------------------------------------
// Inverse DFT along D: SPo[(b,c),16,16,16] -> G1[(b,c),64,16,16] (e^{+i}).
// ---------------------------------------------------------------------------
__global__ __launch_bounds__(256) void idftd_kernel(
    const float* __restrict__ SPo, float* __restrict__ G1) {
  __shared__ float cs[1024], sn[1024];               // [d(64)][kd(16)]
  int t = threadIdx.x;
  for (int i = t; i < 1024; i += 256) {
    int d = i >> 4, kd = i & 15;
    int f = (kd < 8) ? kd : (48 + kd);
    float ang = 2.f * PI_F * (float)(f * d) / 64.f;
    cs[i] = __cosf(ang); sn[i] = __sinf(ang);
  }
  __syncthreads();
  long gid = (long)blockIdx.x * 256 + t;             // 2^20
  int kw = (int)(gid & 7);
  int kh = (int)((gid >> 3) & 15);
  int d  = (int)((gid >> 7) & 63);
  long bc = gid >> 13;
  const float* base = SPo + bc * 4096 + kh * 16 + kw;
  float orr = 0.f, oii = 0.f;
  for (int kd = 0; kd < 16; ++kd) {
    float zr = base[kd * 256], zi = base[kd * 256 + 8];
    float c = cs[d * 16 + kd], si = sn[d * 16 + kd];
    orr += zr * c - zi * si;                         // e^{+i}
    oii += zr * si + zi * c;
  }
  float* ob = G1 + (bc * 64 + d) * 256 + kh * 16 + kw;
  ob[0] = orr; ob[8] = oii;
}

// ---------------------------------------------------------------------------
// Inverse DFT along H: G1[(b,c),64,16,16] -> G2[(b,c,d,h),16]. 4,194,304 thr.
// ---------------------------------------------------------------------------
__global__ __launch_bounds__(256) void idfth_kernel(
    const float* __restrict__ G1, float* __restrict__ G2) {
  __shared__ float cs[1024], sn[1024];               // [h(64)][kh(16)]
  int t = threadIdx.x;
  for (int i = t; i < 1024; i += 256) {
    int h = i >> 4, kh = i & 15;
    int f = (kh < 8) ? kh : (48 + kh);
    float ang = 2.f * PI_F * (float)(f * h) / 64.f;
    cs[i] = __cosf(ang); sn[i] = __sinf(ang);
  }
  __syncthreads();
  long gid = (long)blockIdx.x * 256 + t;             // 2^22
  int kw = (int)(gid & 7);
  int h  = (int)((gid >> 3) & 63);
  int d  = (int)((gid >> 9) & 63);
  long bc = gid >> 15;
  const float* base = G1 + (bc * 64 + d) * 256 + kw;
  __builtin_prefetch(base, 0, 0);
  float orr = 0.f, oii = 0.f;
  for (int kh = 0; kh < 16; ++kh) {
    float zr = base[kh * 16], zi = base[kh * 16 + 8];
    float c = cs[h * 16 + kh], si = sn[h * 16 + kh];
    orr += zr * c - zi * si;
    oii += zr * si + zi * c;
  }
  float* ob = G2 + ((bc * 64 + d) * 64 + h) * 16 + kw;
  ob[0] = orr; ob[8] = oii;
}

// ---------------------------------------------------------------------------
// Inverse rDFT along W (WMMA GEMM): G2[RROWS,16] x IW[16,64] -> Y[RROWS,64].
// IW folds hermitian doubling + full 1/64^3 scale.
// ---------------------------------------------------------------------------
__global__ __launch_bounds__(256) void idftw_wmma(
    const float* __restrict__ G2, float* __restrict__ Y) {
  __shared__ float IW[16 * 64];                      // IW[k*64+w]
  int t = threadIdx.x;
  const float scale = 1.0f / 262144.0f;              // 1/64^3
  for (int i = t; i < 1024; i += 256) {
    int k = i >> 6, w = i & 63;
    float ang = 2.f * PI_F * (float)((k & 7) * w) / 64.f;
    if (k < 8) IW[i] = scale * ((k == 0) ? 1.f : 2.f) * __cosf(ang);
    else       IW[i] = scale * (-2.f) * __sinf(ang); // k=8 -> sin(0)=0
  }
  __syncthreads();
  int lane = t & 31, wv = t >> 5;
  long tile = (long)blockIdx.x * 8 + wv;             // (RROWS/16)*4 = 131072
  long rowtile = tile >> 2;
  int ct = (int)(tile & 3);
  long row0 = rowtile * 16;
  int w0 = ct * 16;
  const float* Arow = G2 + row0 * 16;
  int m = lane & 15;
  int koff = (lane >> 4) << 1;
  int mhi  = (lane >> 4) << 3;
  v8f acc = {0.f,0.f,0.f,0.f,0.f,0.f,0.f,0.f};
  for (int kk = 0; kk < 16; kk += 4) {
    int ka = kk + koff;
    v2f a; a.x = Arow[m * 16 + ka];          a.y = Arow[m * 16 + ka + 1];
    v2f b; b.x = IW[ka * 64 + w0 + m];       b.y = IW[(ka + 1) * 64 + w0 + m];
    acc = wmma_f32_16x16x4(a, b, acc);
  }
  float* Orow = Y + row0 * 64 + w0;
  for (int r = 0; r < 8; ++r) Orow[(long)(r + mhi) * 64 + m] = acc[r];
}

// ---------------------------------------------------------------------------
// Skip channel-mix (WMMA GEMM) fused with spectral add + bias + gelu:
// Xio[b,co,p] = act( sum_ci W[ci,co]*Xin[b,ci,p] + bias[co] + Xio[b,co,p] )
// Wave tile: M=16 (co half), N=16 points, K=32 (8 wmma steps).
// ---------------------------------------------------------------------------
__global__ __launch_bounds__(256) void skip_fuse_wmma(
    const float* __restrict__ Xin, float* __restrict__ Xio,
    const float* __restrict__ skw, const float* __restrict__ skb,
    int do_gelu) {
  __shared__ float Ws[CH * CH];
  __shared__ float Bs[CH];
  int t = threadIdx.x;
  for (int i = t; i < CH * CH; i += 256) Ws[i] = skw[i];
  if (t < CH) Bs[t] = skb[t];
  __syncthreads();
  int lane = t & 31, wv = t >> 5;
  long tile = (long)blockIdx.x * 8 + wv;             // 4*2*16384 = 131072
  int cot = (int)(tile & 1);
  long rest = tile >> 1;
  long pt = rest & 16383;
  int b = (int)(rest >> 14);
  long p0 = pt * 16;
  int co0 = cot * 16;
  int m = lane & 15;
  int koff = (lane >> 4) << 1;
  int mhi  = (lane >> 4) << 3;
  float* Yb = Xio + (long)b * CH * DHW + p0;
  const float* Xb = Xin + (long)b * CH * DHW + p0;
  v8f acc;
  for (int r = 0; r < 8; ++r) {
    int M = r + mhi;
    acc[r] = Yb[(long)(co0 + M) * DHW + m] + Bs[co0 + M];
  }
  for (int kk = 0; kk < CH; kk += 4) {
    int ka = kk + koff;
    v2f a; a.x = Ws[ka * CH + co0 + m];      a.y = Ws[(ka + 1) * CH + co0 + m];
    v2f bb; bb.x = Xb[(long)ka * DHW + m];   bb.y = Xb[(long)(ka + 1) * DHW + m];
    acc = wmma_f32_16x16x4(a, bb, acc);
  }
  for (int r = 0; r < 8; ++r) {
    int M = r + mhi;
    float val = acc[r];
    if (do_gelu) val = gelu_tanh(val);
    Yb[(long)(co0 + M) * DHW + m] = val;
  }
}

// ---------------------------------------------------------------------------
// Spatial mean per (b,c): 128 blocks x 256 threads.
// ---------------------------------------------------------------------------
__global__ __launch_bounds__(256) void mean_kernel(
    const float* __restrict__ X, float* __restrict__ xm) {
  __shared__ float red[256];
  int bc = blockIdx.x, t = threadIdx.x;
  const float* base = X + (long)bc * DHW;
  float s = 0.f;
  for (long i = t; i < DHW; i += 256) s += base[i];
  red[t] = s;
  __syncthreads();
  for (int off = 128; off > 0; off >>= 1) {
    if (t < off) red[t] += red[t + off];
    __syncthreads();
  }
  if (t == 0) xm[bc] = red[0] * (1.0f / (float)DHW);
}

// ---------------------------------------------------------------------------
// Head: proj (linear commutes with mean) -> fc1 relu -> fc2 relu -> fc3.
// ---------------------------------------------------------------------------
__global__ __launch_bounds__(128) void head_kernel(
    const float* __restrict__ xm,
    const float* __restrict__ pw, const float* __restrict__ pb,
    const float* __restrict__ w1, const float* __restrict__ b1,
    const float* __restrict__ w2, const float* __restrict__ b2,
    const float* __restrict__ w3, const float* __restrict__ b3,
    float* __restrict__ out) {
  __shared__ float pm[4 * 32];
  __shared__ float h1[4 * 128];
  __shared__ float h2[4 * 64];
  int t = threadIdx.x;
  {
    int b = t >> 5, co = t & 31;
    float s = pb[co];
    for (int ci = 0; ci < 32; ++ci) s += xm[b * 32 + ci] * pw[ci * 32 + co];
    pm[t] = s;
  }
  __syncthreads();
  for (int i = t; i < 512; i += 128) {
    int b = i >> 7, j = i & 127;
    float s = b1[j];
    for (int c = 0; c < 32; ++c) s += pm[b * 32 + c] * w1[c * 128 + j];
    h1[i] = fmaxf(s, 0.f);
  }
  __syncthreads();
  for (int i = t; i < 256; i += 128) {
    int b = i >> 6, j = i & 63;
    float s = b2[j];
    for (int c = 0; c < 128; ++c) s += h1[b * 128 + c] * w2[c * 64 + j];
    h2[i] = fmaxf(s, 0.f);
  }
  __syncthreads();
  if (t < 12) {
    int b = t / 3, j = t % 3;
    float s = b3[j];
    for (int c = 0; c < 64; ++c) s += h2[b * 64 + c] * w3[c * 3 + j];
    out[b * 3 + j] = s;
  }
}

// ---------------------------------------------------------------------------
extern "C" void kernel_launch(void* const* d_in, const int* in_sizes, int n_in,
                              void* d_out, int out_size, void* d_ws, size_t ws_size,
                              hipStream_t stream) {
  const float* vel = (const float*)d_in[0];
  const float* ptt = (const float*)d_in[1];
  const float* lw  = (const float*)d_in[2];
  const float* lb  = (const float*)d_in[3];
  const float* swr = (const float*)d_in[4];
  const float* swi = (const float*)d_in[5];
  const float* skw = (const float*)d_in[6];
  const float* skb = (const float*)d_in[7];
  const float* pw  = (const float*)d_in[8];
  const float* pb  = (const float*)d_in[9];
  const float* w1  = (const float*)d_in[10];
  const float* b1  = (const float*)d_in[11];
  const float* w2  = (const float*)d_in[12];
  const float* b2  = (const float*)d_in[13];
  const float* w3  = (const float*)d_in[14];
  const float* b3  = (const float*)d_in[15];
  float* out = (float*)d_out;
  float* WS = (float*)d_ws;

  // Workspace layout (floats): two 128MB activations + reused DFT buffers.
  float* XA  = WS;                    // 33,554,432
  float* XB  = WS + 33554432L;        // 33,554,432
  float* F1  = WS + 67108864L;        //  8,388,608  (reused as G2)
  float* F2  = WS + 75497472L;        //  2,097,152  (reused as G1)
  float* SP  = WS + 77594624L;        //    524,288
  float* SPo = WS + 78118912L;        //    524,288
  float* XM  = WS + 78643200L;        //        128

  lift_kernel<<<4096, 256, 0, stream>>>(vel, ptt, lw, lb, XA);

  float* xcur = XA;
  float* xnxt = XB;
  for (int l = 0; l < 4; ++l) {
    dftw_fwd<<<4096, 256, 0, stream>>>(xcur, F1);
    dfth_fwd<<<4096, 256, 0, stream>>>(F1, F2);
    dftd_fwd<<<1024, 256, 0, stream>>>(F2, SP);
    spectral_core<<<2048, 128, 0, stream>>>(
        SP, swr + (long)l * 2097152L, swi + (long)l * 2097152L, SPo);
    idftd_kernel<<<4096, 256, 0, stream>>>(SPo, F2);    // G1 = F2
    idfth_kernel<<<16384, 256, 0, stream>>>(F2, F1);    // G2 = F1
    idftw_wmma<<<16384, 256, 0, stream>>>(F1, xnxt);    // spectral real out
    skip_fuse_wmma<<<16384, 256, 0, stream>>>(
        xcur, xnxt, skw + l * 1024, skb + l * 32, (l < 3) ? 1 : 0);
    float* tmp = xcur; xcur = xnxt; xnxt = tmp;
  }

  mean_kernel<<<128, 256, 0, stream>>>(xcur, XM);
  head_kernel<<<1, 128, 0, stream>>>(XM, pw, pb, w1, b1, w2, b2, w3, b3, out);
}